// SegmentDecoder_72834055406374
// MI455X (gfx1250) — compile-verified
//
#include <hip/hip_runtime.h>
#include <hip/hip_bf16.h>

typedef __attribute__((ext_vector_type(2))) float v2f;
typedef __attribute__((ext_vector_type(4))) float v4f;
typedef __attribute__((ext_vector_type(8))) float v8f;

#define TILE 128
#define KC   32
#define LSTR 36   // padded LDS row stride (floats): %4==0 for b128 staging, conflict-free frag reads

__launch_bounds__(256, 1)
__global__ void segdec_gram_tile_kernel(const float* __restrict__ z,
                                        const int* __restrict__ cls,
                                        const int* __restrict__ batch,
                                        float* __restrict__ out,
                                        int N, int D) {
  __shared__ float As[TILE * LSTR];
  __shared__ float Bs[TILE * LSTR];
  __shared__ int rowB[TILE], rowC[TILE], colB[TILE], colC[TILE];

  const int r0 = blockIdx.y * TILE;
  const int c0 = blockIdx.x * TILE;
  const int tid = (int)threadIdx.x;

  // Uniform (block-scalar) batch-range overlap test; batch[] is sorted ascending.
  const int rl = batch[min(r0, N - 1)];
  const int rh = batch[min(r0 + TILE - 1, N - 1)];
  const int cl = batch[min(c0, N - 1)];
  const int ch = batch[min(c0 + TILE - 1, N - 1)];
  const bool overlap = (rh >= cl) && (ch >= rl);

  if (!overlap) {
    // Zero-fill tile: pure streaming-store path (bandwidth bound).
    const int row = r0 + (tid >> 1);
    const int jb  = c0 + (tid & 1) * 64;
    if (row < N) {
      float* __restrict__ o = out + (size_t)row * (size_t)N;
      const v4f z4 = {0.f, 0.f, 0.f, 0.f};
      #pragma unroll
      for (int jj = 0; jj < 64; jj += 4) {
        const int j = jb + jj;
        if (j + 4 <= N) {
          __builtin_nontemporal_store(z4, (v4f*)(o + j));
        } else {
          for (int e = 0; e < 4; ++e)
            if (j + e < N) __builtin_nontemporal_store(0.f, o + j + e);
        }
      }
    }
    return;
  }

  // ---- compute tile (rare: only the sorted-batch diagonal band) ----
  if (tid < TILE) {
    const int i = r0 + tid;
    rowB[tid] = (i < N) ? batch[i] : -1;
    rowC[tid] = (i < N) ? cls[i]   : -1;
    const int j = c0 + tid;
    colB[tid] = (j < N) ? batch[j] : -2;
    colC[tid] = (j < N) ? cls[j]   : -2;
  }

  const int lane  = tid & 31;
  const int wave  = tid >> 5;        // 8 waves
  const int half  = lane >> 4;
  const int l16   = lane & 15;
  const int strip = wave * 16;       // this wave's M offset in the tile

  v8f acc[8];
  #pragma unroll
  for (int n = 0; n < 8; ++n) acc[n] = v8f{0, 0, 0, 0, 0, 0, 0, 0};

  for (int kc = 0; kc < D; kc += KC) {
    __syncthreads();
    // Stage 128x32 K-chunks of row-strip (A) and col-strip (B) via b128 loads/stores.
    #pragma unroll
    for (int i = 0; i < 4; ++i) {
      const int p   = tid + i * 256;     // 1024 v4f slots per matrix
      const int row = p >> 3;            // 8 v4f per row (KC=32)
      const int c4  = (p & 7) * 4;
      const int gr  = r0 + row;
      const v4f va = (gr < N) ? *(const v4f*)(z + (size_t)gr * D + kc + c4)
                              : v4f{0.f, 0.f, 0.f, 0.f};
      *(v4f*)(As + row * LSTR + c4) = va;
      const int gc = c0 + row;
      const v4f vb = (gc < N) ? *(const v4f*)(z + (size_t)gc * D + kc + c4)
                              : v4f{0.f, 0.f, 0.f, 0.f};
      *(v4f*)(Bs + row * LSTR + c4) = vb;
    }
    __syncthreads();

    #pragma unroll
    for (int k = 0; k < KC; k += 4) {
      // A fragment 16x4 (ISA 7.12.2): lanes 0-15 K=k..k+1, lanes 16-31 K=k+2..k+3
      const float* ap = As + (strip + l16) * LSTR + k + 2 * half;
      v2f a; a.x = ap[0]; a.y = ap[1];
      #pragma unroll
      for (int n = 0; n < 8; ++n) {
        // B fragment 4x16: element (K, N=l16) = z[c0 + n*16 + l16][kc+K]
        const float* bp = Bs + (n * 16 + l16) * LSTR + k + 2 * half;
        v2f b; b.x = bp[0]; b.y = bp[1];
        acc[n] = __builtin_amdgcn_wmma_f32_16x16x4_f32(
            false, a, false, b, (short)0, acc[n], false, false);
      }
    }
  }

  // Masked epilogue. C layout: VGPR v -> (M = v + 8*half, N = l16).
  #pragma unroll
  for (int n = 0; n < 8; ++n) {
    const int j  = c0 + n * 16 + l16;
    const int bj = colB[n * 16 + l16];
    const int cj = colC[n * 16 + l16];
    #pragma unroll
    for (int v = 0; v < 8; ++v) {
      const int mi = strip + v + 8 * half;
      const int i  = r0 + mi;
      if (i < N && j < N) {
        const int bi = rowB[mi], ci = rowC[mi];
        const bool valid = !(ci >= 24 && ci <= 26);
        const bool m = (bi == bj) && (ci == cj) && valid && (i != j);
        __builtin_nontemporal_store(m ? acc[n][v] : 0.f,
                                    out + (size_t)i * (size_t)N + j);
      }
    }
  }
}

extern "C" void kernel_launch(void* const* d_in, const int* in_sizes, int n_in,
                              void* d_out, int out_size, void* d_ws, size_t ws_size,
                              hipStream_t stream) {
  const float* z     = (const float*)d_in[0];
  const int*   cls   = (const int*)d_in[1];
  const int*   batch = (const int*)d_in[2];
  float*       out   = (float*)d_out;
  const int N = in_sizes[1];
  const int D = in_sizes[0] / N;   // 256; assumed multiple of KC
  dim3 grid((N + TILE - 1) / TILE, (N + TILE - 1) / TILE);
  segdec_gram_tile_kernel<<<grid, dim3(256), 0, stream>>>(z, cls, batch, out, N, D);
}